// HPSS_42288247996678
// MI455X (gfx1250) — compile-verified
//
#include <hip/hip_runtime.h>
#include <hip/hip_bf16.h>

// HPSS: S (8 images of 513x2048 fp32) -> (S*mask_harm, S*mask_perc)
// harm = median17 along time (contiguous), perc = median17 along freq (strided),
// zero padding participates in the median (matches jnp.pad + sort[k//2]).
//
// Median-of-17 = 9th smallest, computed from 4 sorted quads + 1 singleton:
//   A8 = oddeven_merge(q0,q1), B8 = oddeven_merge(q2,q3)           (9 CEs each)
//   half-cleaner CE(A[i],B[7-i]) -> L = 8 smallest, U = 8 largest  (8 CEs)
//   p = max(L) (8th of 16), q = min(U) (9th of 16), ans = med3(p,x,q)

#define F_DIM    513
#define T_DIM    2048
#define N_IMG    8            // 4*2 leading dims flattened
#define PAD      8            // (K-1)/2, K=17
#define TILE_F   32
#define TILE_T   64
#define LDS_F    (TILE_F + 2 * PAD)   // 48
#define LDS_T    (TILE_T + 2 * PAD)   // 80
#define LDS_N    (LDS_F * LDS_T)      // 3840 floats = 15360 B
#define NTHREADS 256
#define TINY_F32 1.17549435e-38f      // jnp.finfo(float32).tiny

__device__ __forceinline__ void cswap(float& a, float& b) {
    const float lo = fminf(a, b);
    const float hi = fmaxf(a, b);
    a = lo; b = hi;
}

// 5-CE sorting network for 4 elements (ascending)
__device__ __forceinline__ void sort4v(float* v) {
    cswap(v[0], v[1]); cswap(v[2], v[3]);
    cswap(v[0], v[2]); cswap(v[1], v[3]);
    cswap(v[1], v[2]);
}

// Batcher odd-even merge of two sorted quads a,b -> sorted c[8] (9 CEs)
__device__ __forceinline__ void merge44(const float* a, const float* b, float* c) {
    float e0, e1, e2, e3, o0, o1, o2, o3;
    {   // merge evens (a0,a2) with (b0,b2)
        const float m0 = fminf(a[0], b[0]), m1 = fmaxf(a[0], b[0]);
        const float n0 = fminf(a[2], b[2]), n1 = fmaxf(a[2], b[2]);
        e0 = m0; e1 = fminf(n0, m1); e2 = fmaxf(n0, m1); e3 = n1;
    }
    {   // merge odds (a1,a3) with (b1,b3)
        const float m0 = fminf(a[1], b[1]), m1 = fmaxf(a[1], b[1]);
        const float n0 = fminf(a[3], b[3]), n1 = fmaxf(a[3], b[3]);
        o0 = m0; o1 = fminf(n0, m1); o2 = fmaxf(n0, m1); o3 = n1;
    }
    c[0] = e0;
    c[1] = fminf(o0, e1); c[2] = fmaxf(o0, e1);
    c[3] = fminf(o1, e2); c[4] = fmaxf(o1, e2);
    c[5] = fminf(o2, e3); c[6] = fmaxf(o2, e3);
    c[7] = o3;
}

// 9th smallest of {r0,r1,r2,r3 (sorted quads)} U {x}
__device__ __forceinline__ float median17_runs(const float* r0, const float* r1,
                                               const float* r2, const float* r3,
                                               float x) {
    float A[8], B[8];
    merge44(r0, r1, A);
    merge44(r2, r3, B);
    float L[8], U[8];
#pragma unroll
    for (int i = 0; i < 8; ++i) {
        L[i] = fminf(A[i], B[7 - i]);
        U[i] = fmaxf(A[i], B[7 - i]);
    }
    float p = L[0], q = U[0];
#pragma unroll
    for (int i = 1; i < 8; ++i) { p = fmaxf(p, L[i]); q = fminf(q, U[i]); }
#if __has_builtin(__builtin_amdgcn_fmed3f)
    return __builtin_amdgcn_fmed3f(x, p, q);     // v_med3_f32; p <= q
#else
    return fmaxf(p, fminf(x, q));
#endif
}

// median of 17 raw values (w is clobbered); w[16] is the singleton
__device__ __forceinline__ float median17_raw(float* w) {
    sort4v(w); sort4v(w + 4); sort4v(w + 8); sort4v(w + 12);
    return median17_runs(w, w + 4, w + 8, w + 12, w[16]);
}

__global__ __launch_bounds__(NTHREADS) void hpss_kernel(const float* __restrict__ S,
                                                        float* __restrict__ out) {
    __shared__ float tile[LDS_N];

    const int tid = threadIdx.x;
    const int bt  = blockIdx.x;   // time tile
    const int bf  = blockIdx.y;   // freq tile
    const int img = blockIdx.z;   // flattened (4,2)

    const int f0 = bf * TILE_F - PAD;
    const int t0 = bt * TILE_T - PAD;
    const float* __restrict__ Simg = S + (size_t)img * (F_DIM * T_DIM);

    // ---- Stage tile (+halo) into LDS; out-of-range cells -> 0 (zero pad) ----
#pragma unroll
    for (int i = 0; i < LDS_N / NTHREADS; ++i) {   // 15 elements/thread
        const int  li = tid + i * NTHREADS;
        const int  r  = li / LDS_T;
        const int  c  = li - r * LDS_T;
        const int  f  = f0 + r;
        const int  t  = t0 + c;
        const bool in = (f >= 0) & (f < F_DIM) & (t >= 0) & (t < T_DIM);
#if defined(__gfx1250__) && __has_builtin(__builtin_amdgcn_global_load_async_to_lds_b32)
        if (in) {
            // CDNA5 async global->LDS DMA (ASYNCcnt), no VGPR round-trip.
            __builtin_amdgcn_global_load_async_to_lds_b32(
                (int*)(Simg + (size_t)f * T_DIM + t),
                (int*)(&tile[li]),
                /*offset=*/0, /*cpol=*/0);
        } else {
            tile[li] = 0.0f;   // DS store to a disjoint LDS address; no hazard
        }
#else
        tile[li] = in ? Simg[(size_t)f * T_DIM + t] : 0.0f;
#endif
    }
#if defined(__gfx1250__) && __has_builtin(__builtin_amdgcn_global_load_async_to_lds_b32)
#if __has_builtin(__builtin_amdgcn_s_wait_asynccnt)
    __builtin_amdgcn_s_wait_asynccnt(0);
#else
    asm volatile("s_wait_asynccnt 0" ::: "memory");
#endif
#endif
    __syncthreads();

    // ---- Compute: 8 outputs per thread along f (stride 4), fixed t ----
    const int    lt   = tid & (TILE_T - 1);   // 0..63 -> conflict-free LDS, coalesced stores
    const int    lfq  = tid >> 6;             // 0..3  -> window start residue mod 4
    const int    col  = lt + PAD;
    const size_t nTot = (size_t)N_IMG * F_DIM * T_DIM;
    const int    t    = bt * TILE_T + lt;

    // rolling buffer of 4 sorted freq chunks (window rows lfq+4i .. lfq+4i+15)
    float ck[4][4];
#pragma unroll
    for (int m = 0; m < 4; ++m) {
#pragma unroll
        for (int j = 0; j < 4; ++j)
            ck[m][j] = tile[(lfq + 4 * m + j) * LDS_T + col];
        sort4v(ck[m]);
    }

#pragma unroll
    for (int i = 0; i < 8; ++i) {
        const int lf = lfq + 4 * i;
        const int f  = bf * TILE_F + lf;

        // harmonic: median over time window, LDS row lf+PAD, cols lt..lt+16
        float w[17];
        const float* rowp = &tile[(lf + PAD) * LDS_T + lt];
#pragma unroll
        for (int j = 0; j < 17; ++j) w[j] = rowp[j];
        const float s    = w[8];              // S(f,t) = center of the window
        const float harm = median17_raw(w);

        // percussive: median over freq window = chunks i..i+3 + raw row lf+16
        const float x    = tile[(lf + 16) * LDS_T + col];
        const float perc = median17_runs(ck[0], ck[1], ck[2], ck[3], x);

        // softmask with MARGIN=1, POWER=2: shared Z and denominator
        float Z = fmaxf(harm, perc);
        Z       = (Z < TINY_F32) ? 1.0f : Z;
        const float rz = 1.0f / Z;
        float m = harm * rz; m *= m;
        float r = perc * rz; r *= r;
        const float rd = 1.0f / (m + r);

        if (f < F_DIM) {
            const size_t idx = (size_t)img * (F_DIM * T_DIM) + (size_t)f * T_DIM + (size_t)t;
            // streaming outputs: NT hint keeps the L2-resident input un-evicted
            __builtin_nontemporal_store(s * (m * rd), &out[idx]);
            __builtin_nontemporal_store(s * (r * rd), &out[nTot + idx]);
        }

        // slide the freq window: drop chunk i, sort chunk i+4
        if (i < 7) {
#pragma unroll
            for (int mq = 0; mq < 3; ++mq)
#pragma unroll
                for (int j = 0; j < 4; ++j) ck[mq][j] = ck[mq + 1][j];
#pragma unroll
            for (int j = 0; j < 4; ++j)
                ck[3][j] = tile[(lfq + 4 * (i + 4) + j) * LDS_T + col];
            sort4v(ck[3]);
        }
    }
}

extern "C" void kernel_launch(void* const* d_in, const int* in_sizes, int n_in,
                              void* d_out, int out_size, void* d_ws, size_t ws_size,
                              hipStream_t stream) {
    (void)in_sizes; (void)n_in; (void)out_size; (void)d_ws; (void)ws_size;
    const float* S   = (const float*)d_in[0];
    float*       out = (float*)d_out;

    dim3 grid(T_DIM / TILE_T,                       // 32 time tiles
              (F_DIM + TILE_F - 1) / TILE_F,        // 17 freq tiles
              N_IMG);                               // 8 images
    hpss_kernel<<<grid, NTHREADS, 0, stream>>>(S, out);
}